// mnist_node_pred_GNN_12086037971444
// MI455X (gfx1250) — compile-verified
//
#include <hip/hip_runtime.h>
#include <hip/hip_bf16.h>
#include <cstdint>
#include <cstddef>

// ---------------- problem constants (match reference) ----------------
#define N_NODES  10000
#define N_EDGES  160000
#define IN_FEATS 784
#define HID      512
#define NCLS     10
#define NCLS_PAD 16

typedef __attribute__((ext_vector_type(2))) float v2f;
typedef __attribute__((ext_vector_type(8))) float v8f;

// ---------------- degree / norm ----------------
__global__ void k_deg_init(int* __restrict__ cnt) {
    int i = blockIdx.x * blockDim.x + threadIdx.x;
    if (i < N_NODES) cnt[i] = 1;              // self-loop
}

__global__ void k_deg_count(const long long* __restrict__ dst, int* __restrict__ cnt) {
    int e = blockIdx.x * blockDim.x + threadIdx.x;
    if (e < N_EDGES) atomicAdd(&cnt[(int)dst[e]], 1);
}

__global__ void k_dinv(const int* __restrict__ cnt, float* __restrict__ dinv) {
    int i = blockIdx.x * blockDim.x + threadIdx.x;
    if (i < N_NODES) dinv[i] = rsqrtf((float)cnt[i]);
}

// ---------------- CSR build: exclusive scan of in-degree (cnt-1) ----------------
// Single 1024-thread block, LDS Hillis-Steele scan with serial chunk carry.
__global__ __launch_bounds__(1024) void k_scan(const int* __restrict__ cnt,
                                               int* __restrict__ off) {
    __shared__ int carry;
    __shared__ int buf[1024];
    if (threadIdx.x == 0) carry = 0;
    __syncthreads();
    for (int base = 0; base < N_NODES; base += 1024) {
        int i = base + threadIdx.x;
        int v = (i < N_NODES) ? (cnt[i] - 1) : 0;    // real in-degree (no self-loop)
        buf[threadIdx.x] = v;
        __syncthreads();
        for (int s = 1; s < 1024; s <<= 1) {
            int t = (threadIdx.x >= (unsigned)s) ? buf[threadIdx.x - s] : 0;
            __syncthreads();
            buf[threadIdx.x] += t;
            __syncthreads();
        }
        if (i < N_NODES) off[i] = carry + buf[threadIdx.x] - v;  // exclusive
        __syncthreads();
        if (threadIdx.x == 1023) carry += buf[1023];
        __syncthreads();
    }
    if (threadIdx.x == 0) off[N_NODES] = carry;      // == N_EDGES
}

__global__ void k_zero_i32(int* __restrict__ p, int n) {
    int i = blockIdx.x * blockDim.x + threadIdx.x;
    if (i < n) p[i] = 0;
}

__global__ void k_csr_fill(const long long* __restrict__ src,
                           const long long* __restrict__ dst,
                           const int* __restrict__ off, int* __restrict__ fill,
                           int* __restrict__ csr_src) {
    int e = blockIdx.x * blockDim.x + threadIdx.x;
    if (e >= N_EDGES) return;
    int d = (int)dst[e];
    int pos = off[d] + atomicAdd(&fill[d], 1);
    csr_src[pos] = (int)src[e];
}

// ---------------- fp32 WMMA GEMM: C[M x ldC] = A[M x K] * B[K x LDB] ----------------
// One wave computes a 16 x (16*NT) tile using V_WMMA_F32_16X16X4_F32.
// Software-pipelined: next step's A + all NT B tiles loaded into registers
// before the 8-WMMA burst of the current step.
template <int NT, int LDB, bool GUARDN>
__global__ __launch_bounds__(256) void k_wmma_gemm(
    const float* __restrict__ A, const float* __restrict__ B, float* __restrict__ C,
    int M, int K, int Nc, int ldC)
{
    const int wavesPerRow = ldC / (16 * NT);
    const int totalWaves  = (M / 16) * wavesPerRow;
    int wave = blockIdx.x * (blockDim.x >> 5) + (threadIdx.x >> 5);
    if (wave >= totalWaves) return;           // whole-wave exit: EXEC all-1s for WMMA waves

    const int lane = threadIdx.x & 31;
    const int h    = lane >> 4;               // half-wave select
    const int ln   = lane & 15;
    const int mtile = wave / wavesPerRow;
    const int ng    = wave - mtile * wavesPerRow;
    const int n0    = ng * (16 * NT) + ln;

    const float* Ap = A + (size_t)(mtile * 16 + ln) * K + 2 * h;  // += 4 per step
    const float* Bp = B + (size_t)(2 * h) * LDB + n0;             // += 4*LDB per step

    v8f acc[NT] = {};
    v2f a_cur, a_nxt;
    v2f b_cur[NT], b_nxt[NT];

    // ---- prologue: load step 0 ----
    a_cur = *(const v2f*)Ap;
#pragma unroll
    for (int t = 0; t < NT; ++t) {
        if (GUARDN && (n0 + 16 * t) >= Nc) {
            b_cur[t].x = 0.0f; b_cur[t].y = 0.0f;
        } else {
            b_cur[t].x = Bp[16 * t];
            b_cur[t].y = Bp[LDB + 16 * t];
        }
    }

    // ---- steady state: issue loads for k+4, then WMMA burst for k ----
    for (int k = 0; k < K - 4; k += 4) {
        Ap += 4;
        Bp += 4 * LDB;
        a_nxt = *(const v2f*)Ap;
#pragma unroll
        for (int t = 0; t < NT; ++t) {
            if (GUARDN && (n0 + 16 * t) >= Nc) {
                b_nxt[t].x = 0.0f; b_nxt[t].y = 0.0f;
            } else {
                b_nxt[t].x = Bp[16 * t];
                b_nxt[t].y = Bp[LDB + 16 * t];
            }
        }
#pragma unroll
        for (int t = 0; t < NT; ++t)
            acc[t] = __builtin_amdgcn_wmma_f32_16x16x4_f32(
                false, a_cur, false, b_cur[t], (short)0, acc[t], false, false);
        a_cur = a_nxt;
#pragma unroll
        for (int t = 0; t < NT; ++t) b_cur[t] = b_nxt[t];
    }

    // ---- epilogue: last step ----
#pragma unroll
    for (int t = 0; t < NT; ++t)
        acc[t] = __builtin_amdgcn_wmma_f32_16x16x4_f32(
            false, a_cur, false, b_cur[t], (short)0, acc[t], false, false);

    // ---- store ----
#pragma unroll
    for (int t = 0; t < NT; ++t) {
        float* Cr = C + (size_t)(mtile * 16 + 8 * h) * ldC + (n0 + 16 * t);
#pragma unroll
        for (int r = 0; r < 8; ++r)
            Cr[(size_t)r * ldC] = acc[t][r];
    }
}

// ---------------- fused GCN aggregation (CSR gather, no atomics) ----------------
// out[i] = relu?( d_i^2 * G[i] + bias + sum_{e in CSR(i)} d_src*d_i * G[src] )
// One block per dst node; each thread owns 4 consecutive features in registers.
__global__ void k_gather(const float* __restrict__ G, const float* __restrict__ bias,
                         const float* __restrict__ dinv,
                         const int* __restrict__ off, const int* __restrict__ csr_src,
                         float* __restrict__ out,
                         int F, int Fvalid, int do_relu)
{
    const int i   = blockIdx.x;
    const int tid = threadIdx.x;
    const int f   = tid * 4;
    const bool act = f < F;

    const float di = dinv[i];
    float ax = 0.f, ay = 0.f, az = 0.f, aw = 0.f;
    if (act) {
        const float4 v = ((const float4*)(G + (size_t)i * F))[tid];
        const float bx = (f + 0 < Fvalid) ? bias[f + 0] : 0.0f;
        const float by = (f + 1 < Fvalid) ? bias[f + 1] : 0.0f;
        const float bz = (f + 2 < Fvalid) ? bias[f + 2] : 0.0f;
        const float bw = (f + 3 < Fvalid) ? bias[f + 3] : 0.0f;
        const float d2 = di * di;
        ax = d2 * v.x + bx; ay = d2 * v.y + by;
        az = d2 * v.z + bz; aw = d2 * v.w + bw;
    }

    const int beg = off[i], end = off[i + 1];
    for (int idx = beg; idx < end; ++idx) {
        const int   s = csr_src[idx];          // uniform across block (broadcast)
        const float w = dinv[s] * di;
        if (act) {
            const float4 u = ((const float4*)(G + (size_t)s * F))[tid];
            ax += w * u.x; ay += w * u.y; az += w * u.z; aw += w * u.w;
        }
    }

    if (act) {
        if (do_relu) {
            ax = fmaxf(ax, 0.f); ay = fmaxf(ay, 0.f);
            az = fmaxf(az, 0.f); aw = fmaxf(aw, 0.f);
        }
        float4 r; r.x = ax; r.y = ay; r.z = az; r.w = aw;
        ((float4*)(out + (size_t)i * F))[tid] = r;
    }
}

// ---------------- final log-softmax over 10 classes ----------------
__global__ void k_logsoftmax(const float* __restrict__ LA, float* __restrict__ out) {
    int i = blockIdx.x * blockDim.x + threadIdx.x;
    if (i >= N_NODES) return;
    const float* v = LA + (size_t)i * NCLS_PAD;
    float m = v[0];
#pragma unroll
    for (int c = 1; c < NCLS; ++c) m = fmaxf(m, v[c]);
    float s = 0.0f;
#pragma unroll
    for (int c = 0; c < NCLS; ++c) s += expf(v[c] - m);
    float l = m + logf(s);
    float* o = out + (size_t)i * NCLS;
#pragma unroll
    for (int c = 0; c < NCLS; ++c) o[c] = v[c] - l;
}

// ---------------- host launcher ----------------
extern "C" void kernel_launch(void* const* d_in, const int* in_sizes, int n_in,
                              void* d_out, int out_size, void* d_ws, size_t ws_size,
                              hipStream_t stream)
{
    const float*     x   = (const float*)d_in[0];
    const long long* ei  = (const long long*)d_in[1];   // int64 edge_index [2, E]
    const float*     W1  = (const float*)d_in[2];
    const float*     b1  = (const float*)d_in[3];
    const float*     W2  = (const float*)d_in[4];
    const float*     b2  = (const float*)d_in[5];
    const float*     Wc  = (const float*)d_in[6];
    const float*     bc  = (const float*)d_in[7];
    const long long* src = ei;
    const long long* dst = ei + N_EDGES;

    // workspace carve-up (all offsets 16B aligned)
    char*  ws      = (char*)d_ws;
    float* G       = (float*)(ws + 0);            // 10000*512 f32 = 20,480,000 B
    float* AGG     = (float*)(ws + 20480000);     // 10000*512 f32
    float* LG      = (float*)(ws + 40960000);     // 10000*16  f32 = 640,000 B
    float* LA      = (float*)(ws + 41600000);     // 10000*16  f32
    float* dinv    = (float*)(ws + 42240000);     // 10000 f32
    int*   cnt     = (int*)  (ws + 42280000);     // 10000 i32
    int*   off     = (int*)  (ws + 42320000);     // 10001 i32 (padded to 40,016 B)
    int*   fill    = (int*)  (ws + 42360016);     // 10000 i32
    int*   csr_src = (int*)  (ws + 42400016);     // 160000 i32

    dim3 b256(256);

    // ---- graph preprocessing: degrees, norm, CSR ----
    k_deg_init <<<(N_NODES + 255) / 256, b256, 0, stream>>>(cnt);
    k_deg_count<<<(N_EDGES + 255) / 256, b256, 0, stream>>>(dst, cnt);
    k_dinv     <<<(N_NODES + 255) / 256, b256, 0, stream>>>(cnt, dinv);
    k_scan     <<<1, 1024, 0, stream>>>(cnt, off);
    k_zero_i32 <<<(N_NODES + 255) / 256, b256, 0, stream>>>(fill, N_NODES);
    k_csr_fill <<<(N_EDGES + 255) / 256, b256, 0, stream>>>(src, dst, off, fill, csr_src);

    // ---- layer 1: relu(gcn(x, W1, b1)) ----
    {
        int waves = (N_NODES / 16) * (HID / 128);
        k_wmma_gemm<8, HID, false><<<(waves + 7) / 8, b256, 0, stream>>>(
            x, W1, G, N_NODES, IN_FEATS, HID, HID);
    }
    k_gather<<<N_NODES, dim3(128), 0, stream>>>(G, b1, dinv, off, csr_src, AGG,
                                                HID, HID, /*relu=*/1);

    // ---- layer 2: relu(gcn(h1, W2, b2)) ----
    {
        int waves = (N_NODES / 16) * (HID / 128);
        k_wmma_gemm<8, HID, false><<<(waves + 7) / 8, b256, 0, stream>>>(
            AGG, W2, G, N_NODES, HID, HID, HID);
    }
    k_gather<<<N_NODES, dim3(128), 0, stream>>>(G, b2, dinv, off, csr_src, AGG,
                                                HID, HID, /*relu=*/1);

    // ---- layer 3: gcn(h2, Wc, bc) then log_softmax ----
    {
        int waves = (N_NODES / 16);  // ldC = 16 pad, NT = 1
        k_wmma_gemm<1, NCLS, true><<<(waves + 7) / 8, b256, 0, stream>>>(
            AGG, Wc, LG, N_NODES, HID, NCLS, NCLS_PAD);
    }
    k_gather<<<N_NODES, dim3(32), 0, stream>>>(LG, bc, dinv, off, csr_src, LA,
                                               NCLS_PAD, NCLS, /*relu=*/0);
    k_logsoftmax<<<(N_NODES + 255) / 256, b256, 0, stream>>>(LA, (float*)d_out);
}